// SoftRAMAttentionV2_2559800508424
// MI455X (gfx1250) — compile-verified
//
#include <hip/hip_runtime.h>

#define S_DIM 512
#define B_DIM 64
#define H_DIM 8
#define K_DIM 12
#define P_DIM 9
#define T_DIM 4096
#define THRESH_V 4
#define KP 80      // padded GEMM K: 64 token cols + 9 relbit cols + 7 zero pad
#define NCOL 1536  // 3 * H*B  (Aq | Ak | Ar column blocks)
#define HB 512     // H*B

typedef __attribute__((ext_vector_type(2))) float v2f;
typedef __attribute__((ext_vector_type(8))) float v8f;

// ---------------------------------------------------------------------------
// Kernel 1: build X' (512x80), weight matrix W (80x1536), zero the vote buffer.
// Each (h,b) thread owns its 3 columns of W -> no atomics needed.
// ---------------------------------------------------------------------------
__global__ void prep_kernel(const int* __restrict__ tokens,
                            const int* __restrict__ conn,
                            float* __restrict__ xp,
                            float* __restrict__ w,
                            int* __restrict__ votes) {
    int tid = blockIdx.x * blockDim.x + threadIdx.x;
    if (tid < S_DIM * KP) {
        int i = tid / KP, c = tid % KP;
        float v = 0.0f;
        if (c < B_DIM)              v = (float)tokens[i * B_DIM + c];
        else if (c < B_DIM + P_DIM) v = (float)((i >> (c - B_DIM)) & 1);
        xp[i * KP + c] = v;
    } else if (tid < S_DIM * KP + HB) {
        int hb = tid - S_DIM * KP;
        for (int c = 0; c < KP; ++c) {
            w[c * NCOL + hb]        = 0.0f;
            w[c * NCOL + 512 + hb]  = 0.0f;
            w[c * NCOL + 1024 + hb] = 0.0f;
        }
        for (int k = 0; k < K_DIM; ++k) {
            int c = conn[hb * K_DIM + k];
            float wt = (float)(1 << k);
            int row, col;
            if (c < B_DIM)          { row = c;                       col = hb; }
            else if (c < 2 * B_DIM) { row = c - B_DIM;               col = 512 + hb; }
            else                    { row = B_DIM + (c - 2 * B_DIM); col = 1024 + hb; }
            w[row * NCOL + col] += wt;   // same-thread RMW on owned column: race-free
        }
    } else if (tid < S_DIM * KP + HB + S_DIM * B_DIM) {
        votes[tid - S_DIM * KP - HB] = 0;
    }
}

// ---------------------------------------------------------------------------
// Kernel 2: A = X' @ W via V_WMMA_F32_16X16X4_F32. One wave per 16x16 tile.
// 32 M-tiles x 96 N-tiles = 3072 waves. Integer results <= 4095, exact in f32.
// ---------------------------------------------------------------------------
__global__ void gemm_kernel(const float* __restrict__ xp,
                            const float* __restrict__ w,
                            unsigned short* __restrict__ aout) {
    int wave = (blockIdx.x * blockDim.x + threadIdx.x) >> 5;
    int lane = threadIdx.x & 31;
    int half = lane >> 4;    // 0: lanes 0-15, 1: lanes 16-31
    int l    = lane & 15;
    int mt = wave & 31;      // 32 M tiles
    int nt = wave >> 5;      // 96 N tiles

    v8f acc = {};
    const float* xrow = xp + (mt * 16 + l) * KP + 2 * half;  // A: K = kk + 2*half + {0,1}
    const float* wcol = w + nt * 16 + l;                     // B: K = kk + 2*half + {0,1}
    for (int kk = 0; kk < KP; kk += 4) {
        v2f a, b;
        a[0] = xrow[kk];
        a[1] = xrow[kk + 1];
        b[0] = wcol[(kk + 2 * half) * NCOL];
        b[1] = wcol[(kk + 2 * half + 1) * NCOL];
        acc = __builtin_amdgcn_wmma_f32_16x16x4_f32(
            /*neg_a=*/false, a, /*neg_b=*/false, b,
            /*c_mod=*/(short)0, acc, /*reuse_a=*/false, /*reuse_b=*/false);
    }
    int col = nt * 16 + l;
#pragma unroll
    for (int r = 0; r < 8; ++r) {
        int row = mt * 16 + half * 8 + r;   // C/D layout: VGPR r -> M = r + 8*half
        aout[row * NCOL + col] = (unsigned short)(int)(acc[r] + 0.5f);
    }
}

// ---------------------------------------------------------------------------
// Kernel 3: one workgroup per (h,b). ram packed to a 128-dword LDS bitset;
// parity of row i = XOR_{j<=i} bit(Aq[i] + Ak[j] + Ar[i-j]) (disjoint bit
// fields -> plain add). Thread t handles rows t and 511-t: 513 lookups each.
// ---------------------------------------------------------------------------
__global__ void parity_kernel(const float* __restrict__ ram,
                              const unsigned short* __restrict__ aout,
                              int* __restrict__ votes) {
    __shared__ unsigned int  bitset[T_DIM / 32];  // 512 B
    __shared__ unsigned short aq[S_DIM], ak[S_DIM], ar[S_DIM];

    int hb = blockIdx.x;
    int t  = threadIdx.x;  // 0..255

    if (t < T_DIM / 32) {
        unsigned int bits = 0;
        const float* rp = ram + (size_t)hb * T_DIM + t * 32;
        for (int g = 0; g < 32; ++g)
            bits |= (rp[g] > 0.5f ? 1u : 0u) << g;
        bitset[t] = bits;
    }
    for (int r = t; r < S_DIM; r += 256) {
        aq[r] = aout[r * NCOL + hb];
        ak[r] = aout[r * NCOL + 512 + hb];
        ar[r] = aout[r * NCOL + 1024 + hb];
    }
    __syncthreads();

    int b = hb & (B_DIM - 1);
    int i1 = t, i2 = S_DIM - 1 - t;
    unsigned int par1 = 0, par2 = 0;
    {
        int base = aq[i1];
        for (int j = 0; j <= i1; ++j) {
            int addr = base + ak[j] + ar[i1 - j];
            par1 ^= bitset[addr >> 5] >> (addr & 31);
        }
    }
    {
        int base = aq[i2];
        for (int j = 0; j <= i2; ++j) {
            int addr = base + ak[j] + ar[i2 - j];
            par2 ^= bitset[addr >> 5] >> (addr & 31);
        }
    }
    atomicAdd(&votes[i1 * B_DIM + b], (int)(par1 & 1u));
    atomicAdd(&votes[i2 * B_DIM + b], (int)(par2 & 1u));
}

// ---------------------------------------------------------------------------
// Kernel 4: threshold votes -> out (already in (S,B) layout).
// ---------------------------------------------------------------------------
__global__ void final_kernel(const int* __restrict__ votes,
                             float* __restrict__ out) {
    int idx = blockIdx.x * blockDim.x + threadIdx.x;
    if (idx < S_DIM * B_DIM)
        out[idx] = (votes[idx] > THRESH_V) ? 1.0f : 0.0f;
}

extern "C" void kernel_launch(void* const* d_in, const int* in_sizes, int n_in,
                              void* d_out, int out_size, void* d_ws, size_t ws_size,
                              hipStream_t stream) {
    (void)in_sizes; (void)n_in; (void)out_size; (void)ws_size;
    const int*   tokens = (const int*)d_in[0];    // (S,B) int32
    const int*   conn   = (const int*)d_in[1];    // (H,B,K) int32
    const float* ram    = (const float*)d_in[2];  // (H,B,T) float32
    float* out = (float*)d_out;                   // (S,B) float32

    char* wsb = (char*)d_ws;
    float*          xp    = (float*)(wsb);                    // 512*80*4      = 163840 B
    float*          w     = (float*)(wsb + 163840);           // 80*1536*4     = 491520 B
    unsigned short* aout  = (unsigned short*)(wsb + 655360);  // 512*1536*2    = 1572864 B
    int*            votes = (int*)(wsb + 2228224);            // 512*64*4      = 131072 B

    prep_kernel  <<<290, 256, 0, stream>>>(tokens, conn, xp, w, votes); // 74240 threads
    gemm_kernel  <<<384, 256, 0, stream>>>(xp, w, aout);                // 3072 waves
    parity_kernel<<<512, 256, 0, stream>>>(ram, aout, votes);           // one WG per (h,b)
    final_kernel <<<128, 256, 0, stream>>>(votes, out);
}